// DifferentiableLinearSpline_1236950581707
// MI455X (gfx1250) — compile-verified
//
#include <hip/hip_runtime.h>
#include <stdint.h>

// ---------------------------------------------------------------------------
// DifferentiableLinearSpline for MI455X (gfx1250, wave32)
//
// 64 knots x 128 dims staged into LDS via async global->LDS (ASYNCcnt path),
// one wave per query (t uniform per wave), 6-step upper_bound over LDS times,
// float4 lerp, non-temporal coalesced stores (256 MB write-once stream).
// Bandwidth roofline: ~257 MB total traffic / 23.3 TB/s ~= 11 us; WMMA
// reformulation would inflate FLOPs 32x on the slow f32 matrix path with no
// bandwidth win, so the CDNA5-specific path used here is async data movement.
// ---------------------------------------------------------------------------

typedef float v4f __attribute__((ext_vector_type(4)));
typedef int   i4  __attribute__((vector_size(16)));   // pointee type the async
                                                      // builtin expects (b128)

#define NKNOTS 64
#define DIM    128

#if defined(__has_builtin)
#if __has_builtin(__builtin_amdgcn_global_load_async_to_lds_b128)
#define HAVE_ASYNC_B128 1
#endif
#if __has_builtin(__builtin_amdgcn_s_wait_asynccnt)
#define HAVE_WAIT_ASYNC 1
#endif
#endif

__device__ __forceinline__ void async_copy16(const float* g, float* l) {
#if defined(HAVE_ASYNC_B128)
  // Signature (from hipcc diagnostic): (int4 AS1*, int4 AS3*, imm off, imm cpol)
  __builtin_amdgcn_global_load_async_to_lds_b128(
      (__attribute__((address_space(1))) i4*)(g),
      (__attribute__((address_space(3))) i4*)(l),
      /*offset=*/0, /*cpol=*/0);
#else
  // Fallback: raw CDNA5 asm. vdst = LDS byte address VGPR, vaddr = 64-bit
  // global address pair, saddr = off.
  uint32_t lds_off = (uint32_t)(uintptr_t)(__attribute__((address_space(3))) void*)l;
  uint64_t gaddr   = (uint64_t)(uintptr_t)g;
  asm volatile("global_load_async_to_lds_b128 %0, %1, off"
               :: "v"(lds_off), "v"(gaddr) : "memory");
#endif
}

__device__ __forceinline__ void wait_async_all() {
#if defined(HAVE_WAIT_ASYNC)
  __builtin_amdgcn_s_wait_asynccnt(0);
#else
  asm volatile("s_wait_asynccnt 0" ::: "memory");
#endif
}

__global__ void __launch_bounds__(256)
spline_lerp_kernel(const float* __restrict__ t_arr,
                   const float* __restrict__ fixed_points,   // 2 x 128
                   const float* __restrict__ fixed_times,    // 2
                   const float* __restrict__ control_points, // 62 x 128
                   const float* __restrict__ control_times,  // 62
                   float* __restrict__ val_out,              // B x 128
                   float* __restrict__ der_out,              // B x 128
                   int B) {
  __shared__ __align__(16) float s_points[NKNOTS * DIM]; // 32 KB, knot-major
  __shared__ __align__(16) float s_times[NKNOTS + 8];    // padded for search

  const int tid = threadIdx.x; // 256 threads

  // ---- stage knot times (tiny, plain loads) ----
  if (tid < 62)               s_times[1 + tid] = control_times[tid];
  else if (tid == 62)         s_times[0]       = fixed_times[0];
  else if (tid == 63)         s_times[NKNOTS - 1] = fixed_times[1];
  else if (tid < 64 + 8)      s_times[NKNOTS + (tid - 64)] = 3.4e38f; // +inf pad

  // ---- stage knot points via async global->LDS (ASYNCcnt) ----
  // knots 1..62 <- control_points (1984 float4s)
  for (int i = tid; i < (62 * DIM) / 4; i += 256)
    async_copy16(control_points + i * 4, &s_points[DIM + i * 4]);
  // knot 0 <- fixed_points row 0, knot 63 <- fixed_points row 1
  if (tid < 32)
    async_copy16(fixed_points + tid * 4, &s_points[tid * 4]);
  else if (tid < 64)
    async_copy16(fixed_points + DIM + (tid - 32) * 4,
                 &s_points[(NKNOTS - 1) * DIM + (tid - 32) * 4]);

  wait_async_all();
  __syncthreads();

  // ---- main stream: item i = (query q = i>>5, float4 chunk c = i&31) ----
  const long long total  = (long long)B * 32;
  const long long stride = (long long)gridDim.x * blockDim.x;
  for (long long i = (long long)blockIdx.x * blockDim.x + tid; i < total; i += stride) {
    const int q = (int)(i >> 5);      // wave-uniform (32 aligned lanes share q)
    const int c = (int)(i & 31);      // lane's float4 chunk of the 128 dims
    const float tq = t_arr[q];

    // upper_bound(s_times, tq): right = count of times <= tq
    int lo = 0, hi = NKNOTS;
#pragma unroll
    for (int s = 0; s < 6; ++s) {
      const int mid = (lo + hi) >> 1;
      if (s_times[mid] <= tq) lo = mid + 1; else hi = mid;
    }
    int right = lo < 1 ? 1 : (lo > NKNOTS - 1 ? NKNOTS - 1 : lo);
    const int left = right - 1;

    const bool below = (tq <= s_times[0]);
    const bool above = (tq >= s_times[NKNOTS - 1]);

    v4f val, der;
    if (below || above) {               // wave-uniform branch
      const int k = below ? 0 : (NKNOTS - 1);
      val = *(const v4f*)&s_points[k * DIM + c * 4];
      der = (v4f){0.f, 0.f, 0.f, 0.f};
    } else {
      const float tl  = s_times[left];
      const float tr  = s_times[right];
      const float inv = 1.0f / (tr - tl);
      const float a   = (tq - tl) * inv;
      const v4f pl = *(const v4f*)&s_points[left  * DIM + c * 4];
      const v4f pr = *(const v4f*)&s_points[right * DIM + c * 4];
      val = (1.0f - a) * pl + a * pr;   // matches reference formula exactly
      der = (pr - pl) * inv;
    }

    const long long o = (long long)q * DIM + (long long)c * 4;
    __builtin_nontemporal_store(val, (v4f*)(val_out + o)); // write-once stream:
    __builtin_nontemporal_store(der, (v4f*)(der_out + o)); // bypass L2 residency
  }
}

extern "C" void kernel_launch(void* const* d_in, const int* in_sizes, int n_in,
                              void* d_out, int out_size, void* d_ws, size_t ws_size,
                              hipStream_t stream) {
  const float* t              = (const float*)d_in[0];
  const float* fixed_points   = (const float*)d_in[1];
  const float* fixed_times    = (const float*)d_in[2];
  const float* control_points = (const float*)d_in[3];
  const float* control_times  = (const float*)d_in[4];
  const int B = in_sizes[0];

  float* val_out = (float*)d_out;
  float* der_out = (float*)d_out + (size_t)B * DIM;

  // 2048 blocks x 256 threads: each thread handles ~16 float4 items via
  // grid-stride; keeps per-block LDS staging traffic (33 KB, L2-resident)
  // negligible vs the 256 MB output stream.
  spline_lerp_kernel<<<2048, 256, 0, stream>>>(
      t, fixed_points, fixed_times, control_points, control_times,
      val_out, der_out, B);
}